// cut_MPNN_13778255085999
// MI455X (gfx1250) — compile-verified
//
#include <hip/hip_runtime.h>
#include <float.h>

// CDNA5 / gfx1250 implementation of the cut-MPNN forward pass.
// Memory-bound workload (edge scatter dominates; h/agg are L2-resident on the
// 192MB L2) -> keep fp32 end-to-end and run all [N,64]x[64,64] GEMMs through
// V_WMMA_F32_16X16X4_F32 (wave32 WMMA). Weight staging uses the CDNA5 async
// global->LDS path (GLOBAL_LOAD_ASYNC_TO_LDS_B128 + s_wait_asynccnt).

constexpr int   HD     = 64;    // hidden width
constexpr int   WSTR   = 68;    // LDS stride for weights (16B-aligned rows for async b128)
constexpr int   ZSTR   = 66;    // LDS stride for z tiles (even -> 8B aligned pairs)
constexpr float BN_EPS = 1e-5f;

typedef __attribute__((ext_vector_type(2))) float v2f;
typedef __attribute__((ext_vector_type(8))) float v8f;
typedef __attribute__((ext_vector_type(4))) int   v4i;

#if defined(__has_builtin)
#if __has_builtin(__builtin_amdgcn_global_load_async_to_lds_b128) && \
    __has_builtin(__builtin_amdgcn_s_wait_asynccnt)
#define USE_ASYNC_LDS 1
#endif
#endif
#ifndef USE_ASYNC_LDS
#define USE_ASYNC_LDS 0
#endif

#if USE_ASYNC_LDS
__device__ __forceinline__ void lds_async_b128(const float* g, float* l) {
  // builtin parameter type is int4* (b128 payload)
  __builtin_amdgcn_global_load_async_to_lds_b128((v4i*)g, (v4i*)l, 0, 0);
}
#endif

__device__ __forceinline__ v8f wmma4(v2f a, v2f b, v8f c) {
  // D = A(16x4 f32) x B(4x16 f32) + C(16x16 f32)
  return __builtin_amdgcn_wmma_f32_16x16x4_f32(false, a, false, b, (short)0, c,
                                               false, false);
}

__device__ __forceinline__ void atomicMaxF(float* addr, float val) {
  unsigned int* ua = (unsigned int*)addr;
  unsigned int cur = __float_as_uint(*addr);
  while (__uint_as_float(cur) < val) {
    unsigned int prev = atomicCAS(ua, cur, __float_as_uint(val));
    if (prev == cur) break;
    cur = prev;
  }
}
__device__ __forceinline__ void atomicMinF(float* addr, float val) {
  unsigned int* ua = (unsigned int*)addr;
  unsigned int cur = __float_as_uint(*addr);
  while (__uint_as_float(cur) > val) {
    unsigned int prev = atomicCAS(ua, cur, __float_as_uint(val));
    if (prev == cur) break;
    cur = prev;
  }
}

// ---------------- WMMA GEMM kernels ----------------

// out = BN( relu( relu((agg + h) @ W1 + b1) @ W2 + b2 ) )   (GIN MLP, fused;
// the self-term "+h" is folded into the A-fragment load so no copy pass is
// needed and agg can be zero-initialized before the edge scatter.)
__global__ __launch_bounds__(256) void k_gin_mlp(
    const float* __restrict__ in, const float* __restrict__ selfh,
    float* __restrict__ out,
    const float* __restrict__ W1, const float* __restrict__ b1,
    const float* __restrict__ W2, const float* __restrict__ b2,
    const float* __restrict__ bnp, int n) {
  __shared__ float sW1[HD * WSTR];
  __shared__ float sW2[HD * WSTR];
  __shared__ float sZ[8 * 16 * ZSTR];

#if USE_ASYNC_LDS
  for (int i = threadIdx.x; i < 1024; i += 256) {  // 64 rows x 16 b128 chunks
    int r = i >> 4, ch = (i & 15) << 2;
    lds_async_b128(W1 + r * HD + ch, sW1 + r * WSTR + ch);
    lds_async_b128(W2 + r * HD + ch, sW2 + r * WSTR + ch);
  }
  __builtin_amdgcn_s_wait_asynccnt(0);
#else
  for (int i = threadIdx.x; i < HD * HD; i += 256) {
    int r = i >> 6, c = i & 63;
    sW1[r * WSTR + c] = W1[i];
    sW2[r * WSTR + c] = W2[i];
  }
#endif
  __syncthreads();

  const int wv = threadIdx.x >> 5;
  const int lane = threadIdx.x & 31;
  const int lm = lane & 15;   // M (and N-within-tile) index
  const int lh = lane >> 4;   // half-wave select
  const int r0 = blockIdx.x * 128 + wv * 16;
  if (r0 >= n) return;        // wave-uniform: EXEC stays all-ones for WMMA

  float* z = sZ + wv * 16 * ZSTR;
  const v8f vzero = {};
  v8f acc[4];
#pragma unroll
  for (int cb = 0; cb < 4; ++cb) acc[cb] = vzero;

  const float* arow = in + (size_t)(r0 + lm) * HD;
  const float* hrow = selfh + (size_t)(r0 + lm) * HD;
#pragma unroll 4
  for (int kk = 0; kk < 16; ++kk) {
    const int k0 = kk * 4 + lh * 2;
    v2f a = *(const v2f*)(arow + k0);
    v2f ah = *(const v2f*)(hrow + k0);
    a = a + ah;  // GIN self-term
#pragma unroll
    for (int cb = 0; cb < 4; ++cb) {
      v2f b;
      b[0] = sW1[(k0    ) * WSTR + cb * 16 + lm];
      b[1] = sW1[(k0 + 1) * WSTR + cb * 16 + lm];
      acc[cb] = wmma4(a, b, acc[cb]);
    }
  }
#pragma unroll
  for (int cb = 0; cb < 4; ++cb) {
    const int c = cb * 16 + lm;
    const float bias = b1[c];
#pragma unroll
    for (int r = 0; r < 8; ++r) {
      float v = acc[cb][r] + bias;
      z[(r + 8 * lh) * ZSTR + c] = v > 0.f ? v : 0.f;
    }
  }
  // GEMM2: reads this wave's private z region (DScnt RAW ordering within wave)
#pragma unroll
  for (int cb = 0; cb < 4; ++cb) acc[cb] = vzero;
#pragma unroll 4
  for (int kk = 0; kk < 16; ++kk) {
    const int k0 = kk * 4 + lh * 2;
    v2f a = *(const v2f*)(z + lm * ZSTR + k0);
#pragma unroll
    for (int cb = 0; cb < 4; ++cb) {
      v2f b;
      b[0] = sW2[(k0    ) * WSTR + cb * 16 + lm];
      b[1] = sW2[(k0 + 1) * WSTR + cb * 16 + lm];
      acc[cb] = wmma4(a, b, acc[cb]);
    }
  }
#pragma unroll
  for (int cb = 0; cb < 4; ++cb) {
    const int c = cb * 16 + lm;
    const float bias = b2[c];
    const float g = bnp[c], be = bnp[HD + c];
    const float mu = bnp[2 * HD + c];
    const float iv = rsqrtf(bnp[3 * HD + c] + BN_EPS);
#pragma unroll
    for (int r = 0; r < 8; ++r) {
      float v = acc[cb][r] + bias;
      v = v > 0.f ? v : 0.f;
      out[(size_t)(r0 + r + 8 * lh) * HD + c] = (v - mu) * g * iv + be;
    }
  }
}

// First GIN layer: z = relu(outer(aggx, w1) + b1) then WMMA GEMM2 + BN
__global__ __launch_bounds__(256) void k_gin1(
    const float* __restrict__ aggx, float* __restrict__ out,
    const float* __restrict__ w1, const float* __restrict__ b1,
    const float* __restrict__ W2, const float* __restrict__ b2,
    const float* __restrict__ bnp, int n) {
  __shared__ float sW2[HD * WSTR];
  __shared__ float sZ[8 * 16 * ZSTR];
#if USE_ASYNC_LDS
  for (int i = threadIdx.x; i < 1024; i += 256) {
    int r = i >> 4, ch = (i & 15) << 2;
    lds_async_b128(W2 + r * HD + ch, sW2 + r * WSTR + ch);
  }
  __builtin_amdgcn_s_wait_asynccnt(0);
#else
  for (int i = threadIdx.x; i < HD * HD; i += 256)
    sW2[(i >> 6) * WSTR + (i & 63)] = W2[i];
#endif
  __syncthreads();

  const int wv = threadIdx.x >> 5;
  const int lane = threadIdx.x & 31;
  const int lm = lane & 15;
  const int lh = lane >> 4;
  const int r0 = blockIdx.x * 128 + wv * 16;
  if (r0 >= n) return;

  float* z = sZ + wv * 16 * ZSTR;
  for (int i = lane; i < 16 * HD; i += 32) {
    int r = i >> 6, c = i & 63;
    float v = aggx[r0 + r] * w1[c] + b1[c];
    z[r * ZSTR + c] = v > 0.f ? v : 0.f;
  }

  const v8f vzero = {};
  v8f acc[4];
#pragma unroll
  for (int cb = 0; cb < 4; ++cb) acc[cb] = vzero;
#pragma unroll 4
  for (int kk = 0; kk < 16; ++kk) {
    const int k0 = kk * 4 + lh * 2;
    v2f a = *(const v2f*)(z + lm * ZSTR + k0);
#pragma unroll
    for (int cb = 0; cb < 4; ++cb) {
      v2f b;
      b[0] = sW2[(k0    ) * WSTR + cb * 16 + lm];
      b[1] = sW2[(k0 + 1) * WSTR + cb * 16 + lm];
      acc[cb] = wmma4(a, b, acc[cb]);
    }
  }
#pragma unroll
  for (int cb = 0; cb < 4; ++cb) {
    const int c = cb * 16 + lm;
    const float bias = b2[c];
    const float g = bnp[c], be = bnp[HD + c];
    const float mu = bnp[2 * HD + c];
    const float iv = rsqrtf(bnp[3 * HD + c] + BN_EPS);
#pragma unroll
    for (int r = 0; r < 8; ++r) {
      float v = acc[cb][r] + bias;
      v = v > 0.f ? v : 0.f;
      out[(size_t)(r0 + r + 8 * lh) * HD + c] = (v - mu) * g * iv + be;
    }
  }
}

// out = leaky_relu(in @ W1 + b1)   (lin1)
__global__ __launch_bounds__(256) void k_lin1(
    const float* __restrict__ in, float* __restrict__ out,
    const float* __restrict__ W1, const float* __restrict__ b1, int n) {
  __shared__ float sW1[HD * WSTR];
#if USE_ASYNC_LDS
  for (int i = threadIdx.x; i < 1024; i += 256) {
    int r = i >> 4, ch = (i & 15) << 2;
    lds_async_b128(W1 + r * HD + ch, sW1 + r * WSTR + ch);
  }
  __builtin_amdgcn_s_wait_asynccnt(0);
#else
  for (int i = threadIdx.x; i < HD * HD; i += 256)
    sW1[(i >> 6) * WSTR + (i & 63)] = W1[i];
#endif
  __syncthreads();

  const int wv = threadIdx.x >> 5;
  const int lane = threadIdx.x & 31;
  const int lm = lane & 15;
  const int lh = lane >> 4;
  const int r0 = blockIdx.x * 128 + wv * 16;
  if (r0 >= n) return;

  const v8f vzero = {};
  v8f acc[4];
#pragma unroll
  for (int cb = 0; cb < 4; ++cb) acc[cb] = vzero;
  const float* arow = in + (size_t)(r0 + lm) * HD;
#pragma unroll 4
  for (int kk = 0; kk < 16; ++kk) {
    const int k0 = kk * 4 + lh * 2;
    v2f a = *(const v2f*)(arow + k0);
#pragma unroll
    for (int cb = 0; cb < 4; ++cb) {
      v2f b;
      b[0] = sW1[(k0    ) * WSTR + cb * 16 + lm];
      b[1] = sW1[(k0 + 1) * WSTR + cb * 16 + lm];
      acc[cb] = wmma4(a, b, acc[cb]);
    }
  }
#pragma unroll
  for (int cb = 0; cb < 4; ++cb) {
    const int c = cb * 16 + lm;
    const float bias = b1[c];
#pragma unroll
    for (int r = 0; r < 8; ++r) {
      float v = acc[cb][r] + bias;
      v = v > 0.f ? v : 0.01f * v;
      out[(size_t)(r0 + r + 8 * lh) * HD + c] = v;
    }
  }
}

// ---------------- scatter / elementwise kernels ----------------

__global__ void k_node_init(const float* __restrict__ x, float* msrc, float* macc,
                            float* aggx, float* deg, int n) {
  int i = blockIdx.x * 256 + threadIdx.x;
  if (i >= n) return;
  float ms = fabsf(x[i]) > 0.f ? 1.f : 0.f;
  msrc[i] = ms; macc[i] = ms; aggx[i] = x[i]; deg[i] = 0.f;
}

__global__ void k_init_B(float* bmax, float* bmin, float* tv, float* cut, int b) {
  int i = blockIdx.x * blockDim.x + threadIdx.x;
  if (i >= b) return;
  bmax[i] = -FLT_MAX; bmin[i] = FLT_MAX; tv[i] = 0.f; cut[i] = 0.f;
}

__global__ void k_edge_first(const int* __restrict__ row, const int* __restrict__ col,
                             const float* __restrict__ x, const float* __restrict__ msrc,
                             float* macc, float* aggx, float* deg, int e) {
  int i = blockIdx.x * 256 + threadIdx.x;
  if (i >= e) return;
  int r = row[i], c = col[i];
  atomicAdd(&macc[c], msrc[r]);
  atomicAdd(&aggx[c], x[r]);
  atomicAdd(&deg[r], 1.f);
}

__global__ void k_mask_fin(float* mask, float* msrc, float* macc, int n) {
  int i = blockIdx.x * 256 + threadIdx.x;
  if (i >= n) return;
  float v = macc[i] > 0.f ? 1.f : 0.f;
  mask[i] = v; msrc[i] = v; macc[i] = v;
}

__global__ void k_h_mask(const float* __restrict__ gout, const float* __restrict__ mask,
                         float* h, int nh) {
  int i = blockIdx.x * 256 + threadIdx.x;
  if (i >= nh) return;
  h[i] = gout[i] * mask[i >> 6];
}

__global__ void k_zero(float* dst, int nh) {
  int i = blockIdx.x * 256 + threadIdx.x;
  if (i >= nh) return;
  dst[i] = 0.f;
}

// agg[col[e]][f] += h[row[e]][f]  (one thread per (edge, feature): coalesced)
__global__ void k_edge_agg(const int* __restrict__ row, const int* __restrict__ col,
                           const float* __restrict__ h, float* agg, long long total) {
  long long i = (long long)blockIdx.x * 256 + threadIdx.x;
  if (i >= total) return;
  int e = (int)(i >> 6), f = (int)(i & 63);
  atomicAdd(&agg[(size_t)col[e] * HD + f], h[(size_t)row[e] * HD + f]);
}

__global__ void k_edge_mask(const int* __restrict__ row, const int* __restrict__ col,
                            const float* __restrict__ msrc, float* macc, int e) {
  int i = blockIdx.x * 256 + threadIdx.x;
  if (i >= e) return;
  atomicAdd(&macc[col[i]], msrc[row[i]]);
}

// h = BN_obn( (h + gout) * mask )
__global__ void k_h_update(const float* __restrict__ gout, const float* __restrict__ mask,
                           const float* __restrict__ bnp, float* h, int nh) {
  int i = blockIdx.x * 256 + threadIdx.x;
  if (i >= nh) return;
  int f = i & 63;
  float v = (h[i] + gout[i]) * mask[i >> 6];
  h[i] = (v - bnp[128 + f]) * bnp[f] * rsqrtf(bnp[192 + f] + BN_EPS) + bnp[64 + f];
}

// h = BN2( gout * mask )
__global__ void k_lin1_post(const float* __restrict__ gout, const float* __restrict__ mask,
                            const float* __restrict__ bnp, float* h, int nh) {
  int i = blockIdx.x * 256 + threadIdx.x;
  if (i >= nh) return;
  int f = i & 63;
  float v = gout[i] * mask[i >> 6];
  h[i] = (v - bnp[128 + f]) * bnp[f] * rsqrtf(bnp[192 + f] + BN_EPS) + bnp[64 + f];
}

// h1 = leaky_relu(h @ lin2_w + b) * mask   (wave32 dot-reduce, 1 wave / node)
__global__ void k_lin2(const float* __restrict__ h, const float* __restrict__ w,
                       const float* __restrict__ b, const float* __restrict__ mask,
                       float* h1, int n) {
  int t = blockIdx.x * 256 + threadIdx.x;
  int node = t >> 5, lane = t & 31;
  if (node >= n) return;
  const float* hr = h + (size_t)node * HD;
  float s = hr[lane] * w[lane] + hr[lane + 32] * w[lane + 32];
#pragma unroll
  for (int off = 16; off; off >>= 1) s += __shfl_xor(s, off, 32);
  if (lane == 0) {
    float v = s + b[0];
    v = v > 0.f ? v : 0.01f * v;
    h1[node] = v * mask[node];
  }
}

__global__ void k_seg_mm(const int* __restrict__ batch, const float* __restrict__ h1,
                         const float* __restrict__ deg, float* bmax, float* bmin,
                         float* tv, int n) {
  int i = blockIdx.x * 256 + threadIdx.x;
  if (i >= n) return;
  int bb = batch[i];
  atomicMaxF(&bmax[bb], h1[i]);
  atomicMinF(&bmin[bb], h1[i]);
  atomicAdd(&tv[bb], deg[i]);
}

__global__ void k_final_h(const int* __restrict__ batch, const float* __restrict__ h1,
                          const float* __restrict__ mask, const float* __restrict__ x,
                          const float* __restrict__ bmax, const float* __restrict__ bmin,
                          float* hfin, int n) {
  int i = blockIdx.x * 256 + threadIdx.x;
  if (i >= n) return;
  int bb = batch[i];
  float den = bmax[bb] + 1e-6f - bmin[bb];
  float v = (h1[i] - bmin[bb]) / den;
  float mm = mask[i];
  hfin[i] = v * mm + mm * 1e-6f + x[i];
}

__global__ void k_target(const float* __restrict__ rec, const float* __restrict__ tot,
                         const float* __restrict__ trand, const float* __restrict__ tv,
                         float* targ, float* a, float* diffA, float* dotA, int b) {
  int i = blockIdx.x * blockDim.x + threadIdx.x;
  if (i >= b) return;
  float feas = rec[i] / tot[i];
  float t = (trand[i] * feas * 0.85f + 0.1f) * (tv[i] + 1e-6f);
  targ[i] = t; a[i] = 1.f; diffA[i] = t; dotA[i] = 0.f;
}

__global__ void k_scan_node(const int* __restrict__ batch, const float* __restrict__ hfin,
                            const float* __restrict__ mask, const float* __restrict__ deg,
                            const float* __restrict__ a, float* diffA, float* dotA, int n) {
  int i = blockIdx.x * 256 + threadIdx.x;
  if (i >= n) return;
  int bb = batch[i];
  float keep = (a[bb] * hfin[i] < 1.0f) ? 1.f : 0.f;
  float mm = mask[i], dd = deg[i];
  float xk = hfin[i] * keep * mm;
  float dk = dd * keep * mm;
  float dnk = dd * (1.f - keep) * mm;
  atomicAdd(&diffA[bb], -dnk);
  atomicAdd(&dotA[bb], xk * dk);
}

__global__ void k_scan_upd(const float* __restrict__ targ, float* a,
                           float* diffA, float* dotA, int b) {
  int i = blockIdx.x * blockDim.x + threadIdx.x;
  if (i >= b) return;
  a[i] = diffA[i] / (dotA[i] + 1e-5f);
  diffA[i] = targ[i];
  dotA[i] = 0.f;
}

__global__ void k_probs(const int* __restrict__ batch, const float* __restrict__ hfin,
                        const float* __restrict__ mask, const float* __restrict__ deg,
                        const float* __restrict__ a, float* outp, float* cut, int n) {
  int i = blockIdx.x * 256 + threadIdx.x;
  if (i >= n) return;
  int bb = batch[i];
  float p = a[bb] * hfin[i] * mask[i];
  p = fminf(fmaxf(p, 0.f), 1.f);
  outp[i] = p;
  atomicAdd(&cut[bb], p * deg[i]);
}

__global__ void k_edge_cut(const int* __restrict__ row, const int* __restrict__ col,
                           const int* __restrict__ batch, const float* __restrict__ outp,
                           float* cut, int e) {
  int i = blockIdx.x * 256 + threadIdx.x;
  if (i >= e) return;
  atomicAdd(&cut[batch[row[i]]], -outp[row[i]] * outp[col[i]]);
}

__global__ void k_loss(const float* __restrict__ cut, float* outloss, int b) {
  __shared__ float s[128];
  float acc = 0.f;
  for (int i = threadIdx.x; i < b; i += blockDim.x) acc += cut[i];
  s[threadIdx.x] = acc;
  __syncthreads();
  for (int off = 64; off; off >>= 1) {
    if ((int)threadIdx.x < off) s[threadIdx.x] += s[threadIdx.x + off];
    __syncthreads();
  }
  if (threadIdx.x == 0) *outloss = s[0] / (float)b;
}

// ---------------- host orchestration ----------------

extern "C" void kernel_launch(void* const* d_in, const int* in_sizes, int n_in,
                              void* d_out, int out_size, void* d_ws, size_t ws_size,
                              hipStream_t stream) {
  (void)n_in; (void)out_size; (void)ws_size;
  const int N  = in_sizes[0];
  const int E  = in_sizes[1] / 2;
  const int B  = in_sizes[3];
  const int NL = in_sizes[11] / (HD * HD);
  const int NH = N * HD;

  const float* x     = (const float*)d_in[0];
  const int*   row   = (const int*)d_in[1];
  const int*   col   = row + E;
  const int*   batch = (const int*)d_in[2];
  const float* rec   = (const float*)d_in[3];
  const float* tot   = (const float*)d_in[4];
  const float* trand = (const float*)d_in[5];
  const float* c1w1  = (const float*)d_in[6];
  const float* c1b1  = (const float*)d_in[7];
  const float* c1w2  = (const float*)d_in[8];
  const float* c1b2  = (const float*)d_in[9];
  const float* c1bn  = (const float*)d_in[10];
  const float* cw1   = (const float*)d_in[11];
  const float* cb1   = (const float*)d_in[12];
  const float* cw2   = (const float*)d_in[13];
  const float* cb2   = (const float*)d_in[14];
  const float* cbn   = (const float*)d_in[15];
  const float* obn   = (const float*)d_in[16];
  const float* l1w   = (const float*)d_in[17];
  const float* l1b   = (const float*)d_in[18];
  const float* bn2p  = (const float*)d_in[19];
  const float* l2w   = (const float*)d_in[20];
  const float* l2b   = (const float*)d_in[21];

  float* w = (float*)d_ws;
  float* h    = w; w += NH;
  float* agg  = w; w += NH;
  float* gout = w; w += NH;
  float* aggx = w; w += N;
  float* msrc = w; w += N;
  float* macc = w; w += N;
  float* mask = w; w += N;
  float* deg  = w; w += N;
  float* h1   = w; w += N;
  float* hfin = w; w += N;
  float* bmax = w; w += B;
  float* bmin = w; w += B;
  float* tv   = w; w += B;
  float* targ = w; w += B;
  float* av   = w; w += B;
  float* diffA = w; w += B;
  float* dotA  = w; w += B;
  float* cut   = w; w += B;

  dim3 blk(256);
  dim3 gN((N + 255) / 256);
  dim3 gNH((NH + 255) / 256);
  dim3 gE((E + 255) / 256);
  long long EH = (long long)E * HD;
  dim3 gEH((unsigned)((EH + 255) / 256));
  dim3 gGEMM((N + 127) / 128);
  dim3 gL2((N * 32 + 255) / 256);
  dim3 blkB(128);
  dim3 gB((B + 127) / 128);

  k_node_init<<<gN, blk, 0, stream>>>(x, msrc, macc, aggx, deg, N);
  k_init_B<<<gB, blkB, 0, stream>>>(bmax, bmin, tv, cut, B);
  k_edge_first<<<gE, blk, 0, stream>>>(row, col, x, msrc, macc, aggx, deg, E);
  k_mask_fin<<<gN, blk, 0, stream>>>(mask, msrc, macc, N);
  k_gin1<<<gGEMM, blk, 0, stream>>>(aggx, gout, c1w1, c1b1, c1w2, c1b2, c1bn, N);
  k_h_mask<<<gNH, blk, 0, stream>>>(gout, mask, h, NH);

  for (int i = 0; i < NL; ++i) {
    k_zero<<<gNH, blk, 0, stream>>>(agg, NH);
    k_edge_agg<<<gEH, blk, 0, stream>>>(row, col, h, agg, EH);
    k_gin_mlp<<<gGEMM, blk, 0, stream>>>(agg, h, gout,
                                         cw1 + i * HD * HD, cb1 + i * HD,
                                         cw2 + i * HD * HD, cb2 + i * HD,
                                         cbn + i * 4 * HD, N);
    k_edge_mask<<<gE, blk, 0, stream>>>(row, col, msrc, macc, E);
    k_mask_fin<<<gN, blk, 0, stream>>>(mask, msrc, macc, N);
    k_h_update<<<gNH, blk, 0, stream>>>(gout, mask, obn + i * 4 * HD, h, NH);
  }

  k_lin1<<<gGEMM, blk, 0, stream>>>(h, gout, l1w, l1b, N);
  k_lin1_post<<<gNH, blk, 0, stream>>>(gout, mask, bn2p, h, NH);
  k_lin2<<<gL2, blk, 0, stream>>>(h, l2w, l2b, mask, h1, N);
  k_seg_mm<<<gN, blk, 0, stream>>>(batch, h1, deg, bmax, bmin, tv, N);
  k_final_h<<<gN, blk, 0, stream>>>(batch, h1, mask, x, bmax, bmin, hfin, N);
  k_target<<<gB, blkB, 0, stream>>>(rec, tot, trand, tv, targ, av, diffA, dotA, B);
  for (int it = 0; it < 30; ++it) {
    k_scan_node<<<gN, blk, 0, stream>>>(batch, hfin, mask, deg, av, diffA, dotA, N);
    k_scan_upd<<<gB, blkB, 0, stream>>>(targ, av, diffA, dotA, B);
  }

  float* outp = (float*)d_out;
  k_probs<<<gN, blk, 0, stream>>>(batch, hfin, mask, deg, av, outp, cut, N);
  k_edge_cut<<<gE, blk, 0, stream>>>(row, col, batch, outp, cut, E);
  k_loss<<<dim3(1), blkB, 0, stream>>>(cut, outp + N, B);
}